// DefTet_867583394188
// MI455X (gfx1250) — compile-verified
//
#include <hip/hip_runtime.h>
#include <hip/hip_bf16.h>
#include <math.h>

typedef __attribute__((ext_vector_type(2))) float v2f;
typedef __attribute__((ext_vector_type(8))) float v8f;

#define EPSF 1e-10f
#define N_TETS 262144
#define N_BND 1000
#define N_GT 4096
#define N_SAMP 20
#define N_PRED 20000      /* N_BND * N_SAMP */
#define PRED_TILES 1250   /* N_PRED / 16 */
#define GT_TILES 256      /* N_GT / 16  */
#define AMIPS_SC 20.0f

/* ---- workspace layout (float offsets) ---- */
#define WS_CHAMFER 0
#define WS_ANALYTIC 1
#define WS_NLOSS 2
#define WS_SV 3
#define WS_SV2 4
#define WS_SV3 5
#define WS_SV4 6
#define WS_AMIPS 7
#define WS_NSUM 8                                /* 3 floats */
#define WS_FACES 16                              /* N_BND * 16 */
#define WS_FNORM (WS_FACES + N_BND * 16)         /* N_BND * 3  */
#define WS_PA0   (WS_FNORM + N_BND * 3 + 8)      /* PRED_TILES*32 */
#define WS_PA1   (WS_PA0 + PRED_TILES * 32)
#define WS_PNRM  (WS_PA1 + PRED_TILES * 32)      /* N_PRED */
#define WS_GB0   (WS_PNRM + N_PRED)              /* GT_TILES*32 */
#define WS_GB1   (WS_GB0 + GT_TILES * 32)
#define WS_GNRM  (WS_GB1 + GT_TILES * 32)        /* N_GT */

/* ------------- threefry2x32 (matches JAX) ------------- */
__host__ __device__ inline unsigned tf_rotl(unsigned x, int r) {
  return (x << r) | (x >> (32 - r));
}
__host__ __device__ inline void threefry2x32(unsigned k0, unsigned k1,
                                             unsigned c0, unsigned c1,
                                             unsigned* o0, unsigned* o1) {
  unsigned ks0 = k0, ks1 = k1, ks2 = k0 ^ k1 ^ 0x1BD11BDAu;
  unsigned x0 = c0 + ks0, x1 = c1 + ks1;
#define TFR(R) { x0 += x1; x1 = tf_rotl(x1, R); x1 ^= x0; }
  TFR(13) TFR(15) TFR(26) TFR(6)   x0 += ks1; x1 += ks2 + 1u;
  TFR(17) TFR(29) TFR(16) TFR(24)  x0 += ks2; x1 += ks0 + 2u;
  TFR(13) TFR(15) TFR(26) TFR(6)   x0 += ks0; x1 += ks1 + 3u;
  TFR(17) TFR(29) TFR(16) TFR(24)  x0 += ks1; x1 += ks2 + 4u;
  TFR(13) TFR(15) TFR(26) TFR(6)   x0 += ks2; x1 += ks0 + 5u;
#undef TFR
  *o0 = x0; *o1 = x1;
}
__device__ inline float bits_to_uniform(unsigned b) {
  return __uint_as_float((b >> 9) | 0x3f800000u) - 1.0f;
}

/* ------------- zero accumulators (every launch; graph-safe) ------------- */
__global__ void k_zero(float* ws) {
  if (threadIdx.x < 16) ws[threadIdx.x] = 0.0f;
}

/* ------------- tet gather + volume moments + AMIPS ------------- */
__global__ void k_tets(const float* __restrict__ vpos,
                       const float* __restrict__ invv,
                       const int* __restrict__ tet,
                       float* __restrict__ tetpos_out,
                       float* __restrict__ ws) {
  int t = blockIdx.x * blockDim.x + threadIdx.x;
  float V = 0.f, V2 = 0.f, V3 = 0.f, V4 = 0.f, am = 0.f;
  if (t < N_TETS) {
    int i0 = tet[t * 4 + 0], i1 = tet[t * 4 + 1];
    int i2 = tet[t * 4 + 2], i3 = tet[t * 4 + 3];
    float ax = vpos[i0 * 3 + 0], ay = vpos[i0 * 3 + 1], az = vpos[i0 * 3 + 2];
    float bx = vpos[i1 * 3 + 0], by = vpos[i1 * 3 + 1], bz = vpos[i1 * 3 + 2];
    float cx = vpos[i2 * 3 + 0], cy = vpos[i2 * 3 + 1], cz = vpos[i2 * 3 + 2];
    float dx = vpos[i3 * 3 + 0], dy = vpos[i3 * 3 + 1], dz = vpos[i3 * 3 + 2];
    float* o = tetpos_out + (size_t)t * 12;
    o[0] = ax; o[1] = ay; o[2] = az;
    o[3] = bx; o[4] = by; o[5] = bz;
    o[6] = cx; o[7] = cy; o[8] = cz;
    o[9] = dx; o[10] = dy; o[11] = dz;
    /* V = -det([A-D; B-D; C-D]) / 6 */
    float r0x = ax - dx, r0y = ay - dy, r0z = az - dz;
    float r1x = bx - dx, r1y = by - dy, r1z = bz - dz;
    float r2x = cx - dx, r2y = cy - dy, r2z = cz - dz;
    float det = r0x * (r1y * r2z - r1z * r2y)
              - r0y * (r1x * r2z - r1z * r2x)
              + r0z * (r1x * r2y - r1y * r2x);
    V = -det / 6.0f; V2 = V * V; V3 = V2 * V; V4 = V2 * V2;
    /* AMIPS: J = inv(V) * ([B-A; C-A; D-A] * 20) */
    float o0[3] = {(bx - ax) * AMIPS_SC, (by - ay) * AMIPS_SC, (bz - az) * AMIPS_SC};
    float o1[3] = {(cx - ax) * AMIPS_SC, (cy - ay) * AMIPS_SC, (cz - az) * AMIPS_SC};
    float o2[3] = {(dx - ax) * AMIPS_SC, (dy - ay) * AMIPS_SC, (dz - az) * AMIPS_SC};
    const float* iv = invv + (size_t)t * 9;
    float J[3][3];
    float fro = 0.f;
#pragma unroll
    for (int i = 0; i < 3; i++) {
      float a0 = iv[i * 3 + 0], a1 = iv[i * 3 + 1], a2 = iv[i * 3 + 2];
#pragma unroll
      for (int k = 0; k < 3; k++) {
        float jv = a0 * o0[k] + a1 * o1[k] + a2 * o2[k];
        J[i][k] = jv; fro += jv * jv;
      }
    }
    float dJ = J[0][0] * (J[1][1] * J[2][2] - J[1][2] * J[2][1])
             - J[0][1] * (J[1][0] * J[2][2] - J[1][2] * J[2][0])
             + J[0][2] * (J[1][0] * J[2][1] - J[1][1] * J[2][0]);
    am = fro / (powf(fabsf(dJ), 0.6666667f) + EPSF);
  }
  __shared__ float red[256];
  float vals[5] = {V, V2, V3, V4, am};
  const int ofs[5] = {WS_SV, WS_SV2, WS_SV3, WS_SV4, WS_AMIPS};
#pragma unroll
  for (int q = 0; q < 5; q++) {
    red[threadIdx.x] = vals[q];
    __syncthreads();
    for (int s = 128; s > 0; s >>= 1) {
      if ((int)threadIdx.x < s) red[threadIdx.x] += red[threadIdx.x + s];
      __syncthreads();
    }
    if (threadIdx.x == 0) atomicAdd(&ws[ofs[q]], red[0]);
    __syncthreads();
  }
}

/* ------------- boundary faces: geometry precompute + normals ------------- */
__global__ void k_faces(const float* __restrict__ vpos,
                        const int* __restrict__ bnd,
                        float* __restrict__ ws) {
  int f = blockIdx.x * blockDim.x + threadIdx.x;
  if (f >= N_BND) return;
  int i0 = bnd[f * 3 + 0], i1 = bnd[f * 3 + 1], i2 = bnd[f * 3 + 2];
  float s0x = vpos[i0 * 3 + 0], s0y = vpos[i0 * 3 + 1], s0z = vpos[i0 * 3 + 2];
  float s1x = vpos[i1 * 3 + 0], s1y = vpos[i1 * 3 + 1], s1z = vpos[i1 * 3 + 2];
  float s2x = vpos[i2 * 3 + 0], s2y = vpos[i2 * 3 + 1], s2z = vpos[i2 * 3 + 2];
  float e0x = s1x - s0x, e0y = s1y - s0y, e0z = s1z - s0z;
  float e1x = s2x - s0x, e1y = s2y - s0y, e1z = s2z - s0z;
  float d00 = e0x * e0x + e0y * e0y + e0z * e0z;
  float d01 = e0x * e1x + e0y * e1y + e0z * e1z;
  float d11 = e1x * e1x + e1y * e1y + e1z * e1z;
  float qx = e1x - e0x, qy = e1y - e0y, qz = e1z - e0z;
  float e12 = qx * qx + qy * qy + qz * qz;
  float* fd = ws + WS_FACES + f * 16;
  fd[0] = s0x; fd[1] = s0y; fd[2] = s0z;
  fd[3] = e0x; fd[4] = e0y; fd[5] = e0z;
  fd[6] = e1x; fd[7] = e1y; fd[8] = e1z;
  fd[9] = d00; fd[10] = d01; fd[11] = d11;
  fd[12] = 1.0f / (d00 * d11 - d01 * d01 + EPSF);
  fd[13] = 1.0f / (d00 + EPSF);
  fd[14] = 1.0f / (e12 + EPSF);
  fd[15] = 1.0f / (d11 + EPSF);
  /* normal */
  float nx = e0y * e1z - e0z * e1y;
  float ny = e0z * e1x - e0x * e1z;
  float nz = e0x * e1y - e0y * e1x;
  float nn = sqrtf(nx * nx + ny * ny + nz * nz) + EPSF;
  nx /= nn; ny /= nn; nz /= nn;
  float* fn = ws + WS_FNORM + f * 3;
  fn[0] = nx; fn[1] = ny; fn[2] = nz;
  atomicAdd(&ws[WS_NSUM + 0], nx);
  atomicAdd(&ws[WS_NSUM + 1], ny);
  atomicAdd(&ws[WS_NSUM + 2], nz);
}

/* ------------- normal loss (single block) ------------- */
__global__ void k_nloss(float* __restrict__ ws) {
  __shared__ float nb[3];
  __shared__ float red[256];
  if (threadIdx.x == 0) {
    float mx = ws[WS_NSUM + 0] / (float)N_BND;
    float my = ws[WS_NSUM + 1] / (float)N_BND;
    float mz = ws[WS_NSUM + 2] / (float)N_BND;
    float nn = sqrtf(mx * mx + my * my + mz * mz) + EPSF;
    nb[0] = mx / nn; nb[1] = my / nn; nb[2] = mz / nn;
  }
  __syncthreads();
  float s = 0.f;
  for (int f = threadIdx.x; f < N_BND; f += 256) {
    const float* fn = ws + WS_FNORM + f * 3;
    s += 1.0f - (fn[0] * nb[0] + fn[1] * nb[1] + fn[2] * nb[2]);
  }
  red[threadIdx.x] = s;
  __syncthreads();
  for (int st = 128; st > 0; st >>= 1) {
    if ((int)threadIdx.x < st) red[threadIdx.x] += red[threadIdx.x + st];
    __syncthreads();
  }
  if (threadIdx.x == 0) ws[WS_NLOSS] = red[0] / (float)N_BND;
}

/* ------------- sample pred points; pack WMMA A operands ------------- */
__global__ void k_predgen(float* __restrict__ ws,
                          unsigned ku0, unsigned ku1,
                          unsigned kv0, unsigned kv1) {
  int j = blockIdx.x * blockDim.x + threadIdx.x;
  if (j >= N_PRED) return;
  unsigned i = (j < N_PRED / 2) ? (unsigned)j : (unsigned)(j - N_PRED / 2);
  unsigned a, b;
  threefry2x32(ku0, ku1, i, i + N_PRED / 2, &a, &b);
  float u = bits_to_uniform((j < N_PRED / 2) ? a : b);
  threefry2x32(kv0, kv1, i, i + N_PRED / 2, &a, &b);
  float v = bits_to_uniform((j < N_PRED / 2) ? a : b);
  int f = j / N_SAMP;
  const float* fd = ws + WS_FACES + f * 16;
  float su = sqrtf(u);
  float w1 = su * (1.0f - v), w2 = su * v;   /* w0+w1+w2 == 1 */
  float px = fd[0] + w1 * fd[3] + w2 * fd[6];
  float py = fd[1] + w1 * fd[4] + w2 * fd[7];
  float pz = fd[2] + w1 * fd[5] + w2 * fd[8];
  ws[WS_PNRM + j] = px * px + py * py + pz * pz;
  int tile = j >> 4, li = j & 15;
  float* a0 = ws + WS_PA0 + tile * 32;
  float* a1 = ws + WS_PA1 + tile * 32;
  /* A 16x4 layout: lanes 0-15 hold K0,K1 (x,y); lanes 16-31 hold K2,K3 (z,0) */
  a0[li] = px; a0[16 + li] = pz;
  a1[li] = py; a1[16 + li] = 0.0f;
}

/* ------------- pack WMMA B operands from gt points ------------- */
__global__ void k_gtprep(const float* __restrict__ gt, float* __restrict__ ws) {
  int g = blockIdx.x * blockDim.x + threadIdx.x;
  if (g >= N_GT) return;
  float x = gt[g * 3 + 0], y = gt[g * 3 + 1], z = gt[g * 3 + 2];
  ws[WS_GNRM + g] = x * x + y * y + z * z;
  int tile = g >> 4, li = g & 15;
  float* b0 = ws + WS_GB0 + tile * 32;
  float* b1 = ws + WS_GB1 + tile * 32;
  /* B 4x16 layout: VGPR0 rows K0(lanes 0-15), K2(lanes 16-31); VGPR1 rows K1,K3 */
  b0[li] = x; b0[16 + li] = z;
  b1[li] = y; b1[16 + li] = 0.0f;
}

/* ------------- chamfer: (20000 x 3) x (3 x 4096) via V_WMMA_F32_16X16X4_F32 ------------- */
__global__ void k_chamfer(float* __restrict__ ws) {
  int wave = (blockIdx.x * blockDim.x + threadIdx.x) >> 5;
  int lane = threadIdx.x & 31;
  if (wave >= PRED_TILES) return;   /* wave-uniform: EXEC stays all-ones for WMMA */
  v2f A;
  A.x = ws[WS_PA0 + wave * 32 + lane];
  A.y = ws[WS_PA1 + wave * 32 + lane];
  float pnr[8], minv[8];
  int mbase = wave * 16 + ((lane >> 4) << 3);   /* C VGPR r -> M = r + 8*(lane/16) */
#pragma unroll
  for (int r = 0; r < 8; r++) {
    pnr[r] = ws[WS_PNRM + mbase + r];
    minv[r] = 3.0e38f;
  }
  const float* gb0 = ws + WS_GB0;
  const float* gb1 = ws + WS_GB1;
  const float* gnrm = ws + WS_GNRM;
  for (int t = 0; t < GT_TILES; t++) {
    v2f B;
    B.x = gb0[t * 32 + lane];
    B.y = gb1[t * 32 + lane];
    float gn = gnrm[t * 16 + (lane & 15)];
    v8f c = {};
    c = __builtin_amdgcn_wmma_f32_16x16x4_f32(
        /*neg_a=*/false, A, /*neg_b=*/false, B,
        /*c_mod=*/(short)0, c, /*reuse_a=*/false, /*reuse_b=*/false);
#pragma unroll
    for (int r = 0; r < 8; r++)
      minv[r] = fminf(minv[r], pnr[r] + gn - 2.0f * c[r]);
  }
  /* min over N: columns live across lanes of each 16-lane half */
#pragma unroll
  for (int off = 1; off < 16; off <<= 1)
#pragma unroll
    for (int r = 0; r < 8; r++)
      minv[r] = fminf(minv[r], __shfl_xor(minv[r], off, 32));
  if (lane == 0 || lane == 16) {
    float s = 0.f;
#pragma unroll
    for (int r = 0; r < 8; r++) s += minv[r];
    atomicAdd(&ws[WS_CHAMFER], s);
  }
}

/* ------------- analytic point-to-mesh distance ------------- */
__global__ void k_analytic(const float* __restrict__ gt, float* __restrict__ ws) {
  int g = blockIdx.x * blockDim.x + threadIdx.x;
  bool act = g < N_GT;
  float px = 0.f, py = 0.f, pz = 0.f, best = 3.0e38f;
  if (act) { px = gt[g * 3]; py = gt[g * 3 + 1]; pz = gt[g * 3 + 2]; }
  if (act) {
    for (int f = 0; f < N_BND; f++) {
      const float* fd = ws + WS_FACES + f * 16;
      float dx = px - fd[0], dy = py - fd[1], dz = pz - fd[2];
      float e0x = fd[3], e0y = fd[4], e0z = fd[5];
      float e1x = fd[6], e1y = fd[7], e1z = fd[8];
      float d00 = fd[9], d01 = fd[10], d11 = fd[11];
      float dp0 = dx * e0x + dy * e0y + dz * e0z;
      float dp1 = dx * e1x + dy * e1y + dz * e1z;
      float bu = (d11 * dp0 - d01 * dp1) * fd[12];
      float bv = (d00 * dp1 - d01 * dp0) * fd[12];
      bool inside = (bu >= 0.f) && (bv >= 0.f) && (bu + bv <= 1.f);
      float rx = dx - bu * e0x - bv * e1x;
      float ry = dy - bu * e0y - bv * e1y;
      float rz = dz - bu * e0z - bv * e1z;
      float din = rx * rx + ry * ry + rz * rz;
      /* edge s0-s1 */
      float t1 = fminf(fmaxf(dp0 * fd[13], 0.f), 1.f);
      float ax = dx - t1 * e0x, ay = dy - t1 * e0y, az = dz - t1 * e0z;
      float d2a = ax * ax + ay * ay + az * az;
      /* edge s1-s2 : dd = d - e0, e = e1 - e0 */
      float qx = dx - e0x, qy = dy - e0y, qz = dz - e0z;
      float ex = e1x - e0x, ey = e1y - e0y, ez = e1z - e0z;
      float t2 = fminf(fmaxf((qx * ex + qy * ey + qz * ez) * fd[14], 0.f), 1.f);
      float bx = qx - t2 * ex, by = qy - t2 * ey, bz = qz - t2 * ez;
      float d2b = bx * bx + by * by + bz * bz;
      /* edge s2-s0 : dd = d - e1, e = -e1 */
      float wx = dx - e1x, wy = dy - e1y, wz = dz - e1z;
      float t3 = fminf(fmaxf(-(wx * e1x + wy * e1y + wz * e1z) * fd[15], 0.f), 1.f);
      float cx = wx + t3 * e1x, cy = wy + t3 * e1y, cz = wz + t3 * e1z;
      float d2c = cx * cx + cy * cy + cz * cz;
      float de = fminf(d2a, fminf(d2b, d2c));
      best = fminf(best, inside ? din : de);
    }
  }
  __shared__ float red[256];
  red[threadIdx.x] = act ? best : 0.f;
  __syncthreads();
  for (int s = 128; s > 0; s >>= 1) {
    if ((int)threadIdx.x < s) red[threadIdx.x] += red[threadIdx.x + s];
    __syncthreads();
  }
  if (threadIdx.x == 0) atomicAdd(&ws[WS_ANALYTIC], red[0]);
}

/* ------------- finalize scalars ------------- */
__global__ void k_finalize(const float* __restrict__ ws, float* __restrict__ out) {
  if (threadIdx.x == 0 && blockIdx.x == 0) {
    out[0] = ws[WS_CHAMFER] / (float)N_PRED;
    out[1] = ws[WS_ANALYTIC] / (float)N_GT;
    out[2] = ws[WS_NLOSS];
    float n = (float)N_TETS;
    float s1 = ws[WS_SV], s2 = ws[WS_SV2], s3 = ws[WS_SV3], s4 = ws[WS_SV4];
    float m = s1 / n;
    out[3] = s4 - 4.f * m * s3 + 6.f * m * m * s2 - 4.f * m * m * m * s1 + n * m * m * m * m;
    out[4] = ws[WS_AMIPS] / n;
  }
}

extern "C" void kernel_launch(void* const* d_in, const int* in_sizes, int n_in,
                              void* d_out, int out_size, void* d_ws, size_t ws_size,
                              hipStream_t stream) {
  (void)in_sizes; (void)n_in; (void)out_size; (void)ws_size;
  const float* vpos = (const float*)d_in[0];   /* (1,100000,3) f32 */
  const float* gt   = (const float*)d_in[1];   /* (1,4096,3)   f32 */
  const float* invv = (const float*)d_in[2];   /* (262144,3,3) f32 */
  const int*   tet  = (const int*)d_in[3];     /* (1,262144,4) i32 */
  const int*   bnd  = (const int*)d_in[4];     /* (1,1000,3)   i32 */
  float* out = (float*)d_out;                  /* 5 scalars + tet_pos */
  float* ws  = (float*)d_ws;

  /* reproduce jax.random.split(jax.random.key(42)) on host (threefry2x32) */
  unsigned a0, b0, a1, b1;
  threefry2x32(0u, 42u, 0u, 2u, &a0, &b0);
  threefry2x32(0u, 42u, 1u, 3u, &a1, &b1);
  unsigned ku0 = a0, ku1 = a1;   /* key for u */
  unsigned kv0 = b0, kv1 = b1;   /* key for v */

  k_zero<<<1, 32, 0, stream>>>(ws);
  k_tets<<<N_TETS / 256, 256, 0, stream>>>(vpos, invv, tet, out + 5, ws);
  k_faces<<<(N_BND + 255) / 256, 256, 0, stream>>>(vpos, bnd, ws);
  k_nloss<<<1, 256, 0, stream>>>(ws);
  k_predgen<<<(N_PRED + 255) / 256, 256, 0, stream>>>(ws, ku0, ku1, kv0, kv1);
  k_gtprep<<<(N_GT + 255) / 256, 256, 0, stream>>>(gt, ws);
  k_chamfer<<<(PRED_TILES * 32 + 255) / 256, 256, 0, stream>>>(ws);
  k_analytic<<<(N_GT + 255) / 256, 256, 0, stream>>>(gt, ws);
  k_finalize<<<1, 1, 0, stream>>>(ws, out);
}